// Att_0_layer2_11879879544265
// MI455X (gfx1250) — compile-verified
//
#include <hip/hip_runtime.h>
#include <hip/hip_bf16.h>

typedef __attribute__((ext_vector_type(2))) float v2f;
typedef __attribute__((ext_vector_type(8))) float v8f;
typedef __attribute__((ext_vector_type(4))) unsigned int v4u;
typedef __attribute__((ext_vector_type(8))) int v8i;
typedef __attribute__((ext_vector_type(4))) int v4i;

#define NEG_INF -1e30f

constexpr int Bdim = 64, Ndim = 4, Adim = 1024, Odim = 128, Ddim = 256, Qdim = 256, Hdim = 128;

__device__ __forceinline__ v8f wmma_f32(v2f a, v2f b, v8f c) {
  // 8 args: (neg_a, A, neg_b, B, c_mod, C, reuse_a, reuse_b)
  return __builtin_amdgcn_wmma_f32_16x16x4_f32(false, a, false, b, (short)0, c, false, false);
}

// -----------------------------------------------------------------------------
// Kernel A: MT[b] = (obj_reps[b] (O=128 x D=256) @ W1[:D] (D=256 x H=128))^T
// Stored transposed (H x O) so WMMA B-fragments become contiguous b64 DS reads.
// grid = 64 blocks, 256 threads (8 waves). wave w -> row tile w; 8 col tiles.
// -----------------------------------------------------------------------------
__global__ __launch_bounds__(256) void precompute_M(const float* __restrict__ obj,
                                                    const float* __restrict__ W1,
                                                    float* __restrict__ MT) {
  const int b = blockIdx.x;
  const int tid = threadIdx.x;
  const int wv = tid >> 5, lane = tid & 31, j = lane & 15, hi = lane >> 4;
  const float* Am = obj + (size_t)b * Odim * Ddim;  // 128 x 256 row-major
  float* MTb = MT + (size_t)b * Odim * Hdim;        // transposed: 128(H) x 128(O)

  const int m0 = wv * 16;
  v8f c[8];
#pragma unroll
  for (int i = 0; i < 8; i++) c[i] = (v8f){0.f, 0.f, 0.f, 0.f, 0.f, 0.f, 0.f, 0.f};

  const float* arow = Am + (size_t)(m0 + j) * Ddim + 2 * hi;
#pragma unroll 2
  for (int k0 = 0; k0 < Ddim; k0 += 4) {
    v2f a = *reinterpret_cast<const v2f*>(arow + k0);
    const int kr = k0 + 2 * hi;
#pragma unroll
    for (int ct = 0; ct < 8; ct++) {
      const int col = ct * 16 + j;
      v2f bf;
      bf.x = W1[(size_t)kr * Hdim + col];
      bf.y = W1[(size_t)(kr + 1) * Hdim + col];
      c[ct] = wmma_f32(a, bf, c[ct]);
    }
  }
  // transposed store: MT[h, o] = C[o, h]
#pragma unroll
  for (int ct = 0; ct < 8; ct++)
#pragma unroll
    for (int r = 0; r < 8; r++)
      MTb[(size_t)(ct * 16 + j) * Odim + (m0 + r + 8 * hi)] = c[ct][r];
}

// -----------------------------------------------------------------------------
// Kernel B: qb[b,n,h] = b1[h] + sum_j q[b,n,j] * W1[D+j, h]
// -----------------------------------------------------------------------------
__global__ __launch_bounds__(128) void precompute_qb(const float* __restrict__ q,
                                                     const float* __restrict__ W1,
                                                     const float* __restrict__ b1,
                                                     float* __restrict__ qb) {
  const int bn = blockIdx.x;
  const int h = threadIdx.x;
  const float* qp = q + (size_t)bn * Qdim;
  float acc = b1[h];
#pragma unroll 4
  for (int jq = 0; jq < Qdim; jq++)
    acc += qp[jq] * W1[(size_t)(Ddim + jq) * Hdim + h];
  qb[(size_t)bn * Hdim + h] = acc;
}

// -----------------------------------------------------------------------------
// Kernel C (fused): TDM-stage MT[b] -> LDS, WMMA GEMM + relu + W2 dot,
// masked softmax, then att2 @ att1.
// -----------------------------------------------------------------------------
#define SM_STRIDE 132  // 128 + TDM pad of 4 DWORDs -> conflict-free b64 reads

__global__ __launch_bounds__(256) void att_fused(const float* __restrict__ att1,
                                                 const float* __restrict__ MT,
                                                 const float* __restrict__ qb,
                                                 const float* __restrict__ W2,
                                                 const float* __restrict__ b2p,
                                                 const int* __restrict__ tags,
                                                 const int* __restrict__ tptr,
                                                 float* __restrict__ out) {
  extern __shared__ float smem[];
  float* sMT = smem;                        // 128 rows(H) x 132 (O + pad)
  float* sLog = smem + Hdim * SM_STRIDE;    // 1024 (logits -> att2)
  float* sRed = sLog + Adim;                // 16
  float* sPart = smem;                      // phase-2 alias (128 floats)

  const int bn = blockIdx.x;
  const int b = bn >> 2;  // N = 4
  const int tid = threadIdx.x;
  const int wv = tid >> 5, lane = tid & 31, j = lane & 15, hi = lane >> 4;

  const float* att1_bn = att1 + (size_t)bn * Adim * Odim;
  const int* tags_bn = tags + (size_t)bn * Adim;
  const float* MTb = MT + (size_t)b * Odim * Hdim;
  const float* qb_bn = qb + (size_t)bn * Hdim;

  // ---- TDM: async-load MT[b] (128x128 f32, row stride 128) into LDS with
  //      hardware padding: 4 DWORDs pad after every 128 DWORDs -> stride 132.
  if (wv == 0) {
    const unsigned long long ga = (unsigned long long)(uintptr_t)MTb;
    const unsigned int lds_addr = (unsigned int)(uintptr_t)sMT;  // low 32 = LDS byte offset

    v4u g0;
    g0.x = 1u;                                  // count=1 (valid), no gather
    g0.y = lds_addr;                            // LDS dest (bytes)
    g0.z = (unsigned int)(ga & 0xffffffffu);    // global_addr[31:0]
    g0.w = (unsigned int)((ga >> 32) & 0x1ffffffu) | (2u << 30);  // addr[56:32] | type=2

    v8i g1;
    g1[0] = (int)((2u << 16)      // data_size = 4 bytes
                | (1u << 20)      // pad_enable
                | (6u << 22)      // pad_interval: 2^(6+1)=128 DWORDs
                | (3u << 25));    // pad_amount: 3+1 = 4 DWORDs
    g1[1] = (int)(128u << 16);    // tensor_dim0[15:0]=128 (atomic_barrier_addr=0)
    g1[2] = (int)(128u << 16);    // tensor_dim0[31:16]=0 | tensor_dim1[15:0]=128
    g1[3] = (int)(128u << 16);    // tensor_dim1[31:16]=0 | tile_dim0=128
    g1[4] = (int)128u;            // tile_dim1=128 | tile_dim2=0
    g1[5] = (int)128u;            // tensor_dim0_stride[31:0]=128
    g1[6] = 0;                    // stride0[47:32]=0 | stride1[15:0]=0
    g1[7] = 0;                    // stride1[47:16]=0

    v4i gz4 = (v4i){0, 0, 0, 0};                     // groups 2/3 unused (2D tile)
    v8i gz8 = (v8i){0, 0, 0, 0, 0, 0, 0, 0};         // extra group (clang-23 form)

    __builtin_amdgcn_tensor_load_to_lds(g0, g1, gz4, gz4, gz8, 0);
    __builtin_amdgcn_s_wait_tensorcnt(0);
  }
  __syncthreads();

  // per-lane column constants
  float qv[8], w2v[8];
#pragma unroll
  for (int ct = 0; ct < 8; ct++) {
    const int hc = ct * 16 + j;
    qv[ct] = qb_bn[hc];
    w2v[ct] = W2[hc];
  }
  const float b2v = b2p[0];

  // ---- GEMM att1[A,O] @ M[O,H], fused relu + W2 dot -> logits ----
  for (int rt = 0; rt < 8; rt++) {
    const int a0 = (wv * 8 + rt) * 16;
    v8f c[8];
#pragma unroll
    for (int i = 0; i < 8; i++) c[i] = (v8f){0.f, 0.f, 0.f, 0.f, 0.f, 0.f, 0.f, 0.f};

    const float* arow = att1_bn + (size_t)(a0 + j) * Odim + 2 * hi;
#pragma unroll 2
    for (int k0 = 0; k0 < Odim; k0 += 4) {
      v2f a = *reinterpret_cast<const v2f*>(arow + k0);
      const int kr = k0 + 2 * hi;
#pragma unroll
      for (int ct = 0; ct < 8; ct++) {
        // B[kr,col] , B[kr+1,col] = MT[col, kr], MT[col, kr+1]  (one b64 read)
        v2f bf = *reinterpret_cast<const v2f*>(sMT + (size_t)(ct * 16 + j) * SM_STRIDE + kr);
        c[ct] = wmma_f32(a, bf, c[ct]);
      }
    }

    float s[8];
#pragma unroll
    for (int r = 0; r < 8; r++) s[r] = 0.f;
#pragma unroll
    for (int ct = 0; ct < 8; ct++) {
#pragma unroll
      for (int r = 0; r < 8; r++) {
        float x = c[ct][r] + qv[ct];
        x = x > 0.f ? x : 0.f;
        s[r] += x * w2v[ct];
      }
    }
    // reduce over the 16-lane N dimension (xor 1,2,4,8 keeps hi-halves apart)
#pragma unroll
    for (int r = 0; r < 8; r++) {
      float v = s[r];
      v += __shfl_xor(v, 1, 32);
      v += __shfl_xor(v, 2, 32);
      v += __shfl_xor(v, 4, 32);
      v += __shfl_xor(v, 8, 32);
      if (j == 0) sLog[a0 + r + 8 * hi] = v + b2v;
    }
  }
  __syncthreads();

  // ---- masked, temperature-scaled softmax over A = 1024 ----
  const float invt = 1.0f / (float)tptr[0];
  float lmax = NEG_INF;
  for (int a = tid; a < Adim; a += 256) {
    const float v = (tags_bn[a] > 0) ? sLog[a] * invt : NEG_INF;
    sLog[a] = v;
    lmax = fmaxf(lmax, v);
  }
#pragma unroll
  for (int off = 16; off >= 1; off >>= 1) lmax = fmaxf(lmax, __shfl_xor(lmax, off, 32));
  if (lane == 0) sRed[wv] = lmax;
  __syncthreads();
  float bmax = sRed[0];
#pragma unroll
  for (int i = 1; i < 8; i++) bmax = fmaxf(bmax, sRed[i]);

  float lsum = 0.f;
  for (int a = tid; a < Adim; a += 256) {
    const float e = __expf(sLog[a] - bmax);
    sLog[a] = e;
    lsum += e;
  }
#pragma unroll
  for (int off = 16; off >= 1; off >>= 1) lsum += __shfl_xor(lsum, off, 32);
  if (lane == 0) sRed[8 + wv] = lsum;
  __syncthreads();
  float bsum = 0.f;
#pragma unroll
  for (int i = 0; i < 8; i++) bsum += sRed[8 + i];
  const float invsum = 1.0f / bsum;
  for (int a = tid; a < Adim; a += 256) sLog[a] *= invsum;
  __syncthreads();

  // ---- out[o] = sum_a att2[a] * att1[a,o]  (att1 reread hits L2) ----
  const int o = tid & 127;
  const int half = tid >> 7;
  const float* colp = att1_bn + (size_t)half * 512 * Odim + o;
  const float* att2p = sLog + half * 512;
  float acc = 0.f;
#pragma unroll 4
  for (int a = 0; a < 512; a++) acc += att2p[a] * colp[(size_t)a * Odim];
  if (half == 1) sPart[o] = acc;
  __syncthreads();
  if (half == 0) out[(size_t)bn * Odim + o] = acc + sPart[o];
}

// -----------------------------------------------------------------------------
extern "C" void kernel_launch(void* const* d_in, const int* in_sizes, int n_in,
                              void* d_out, int out_size, void* d_ws, size_t ws_size,
                              hipStream_t stream) {
  const float* q    = (const float*)d_in[0];  // [B,N,1,Q]
  const float* att1 = (const float*)d_in[1];  // [B,N,A,O]
  const float* obj  = (const float*)d_in[2];  // [B,O,D]
  const int*   tags = (const int*)d_in[3];    // [B,N,A]
  const float* W1   = (const float*)d_in[4];  // [D+Q,H]
  const float* b1   = (const float*)d_in[5];  // [H]
  const float* W2   = (const float*)d_in[6];  // [H,1]
  const float* b2   = (const float*)d_in[7];  // [1]
  const int*   t    = (const int*)d_in[8];    // scalar
  float* out = (float*)d_out;                 // [B,N,O]

  float* MT = (float*)d_ws;                           // 64*128*128 f32 = 4 MB
  float* qb = MT + (size_t)Bdim * Odim * Hdim;        // 64*4*128 f32

  precompute_M<<<Bdim, 256, 0, stream>>>(obj, W1, MT);
  precompute_qb<<<Bdim * Ndim, 128, 0, stream>>>(q, W1, b1, qb);

  const size_t smem = (size_t)(Hdim * SM_STRIDE + Adim + 16) * sizeof(float);  // ~71.8 KB
  att_fused<<<Bdim * Ndim, 256, smem, stream>>>(att1, MT, qb, W2, b2, tags, t, out);
}